// LastLayer_77111842832926
// MI455X (gfx1250) — compile-verified
//
#include <hip/hip_runtime.h>
#include <math.h>

// ---------------------------------------------------------------------------
// GVAE-style two-tower GCN pipeline for MI455X (gfx1250, wave32).
// Dense GEMMs use V_WMMA_F32_16X16X4_F32 (native fp32 WMMA).
// SpMM (segment_sum with unsorted rows) uses per-edge waves + f32 atomics.
// Head: concat-GEMM (mean & logstd share phase-2 A fragments) fused with
// reparameterization, writing d_out directly.
// ---------------------------------------------------------------------------

typedef float v2f __attribute__((ext_vector_type(2)));
typedef float v8f __attribute__((ext_vector_type(8)));

#define HDIM 128
#define ALPHA 0.2f

__device__ __forceinline__ v8f wmma_f32_k4(v2f a, v2f b, v8f c) {
    // 8 args: (neg_a, A, neg_b, B, c_mod, C, reuse_a, reuse_b)
    return __builtin_amdgcn_wmma_f32_16x16x4_f32(false, a, false, b,
                                                 (short)0, c, false, false);
}

// ---------------------------------------------------------------------------
__global__ __launch_bounds__(256) void zero_kernel(float4* __restrict__ p, long n4) {
    long i = (long)blockIdx.x * blockDim.x + threadIdx.x;
    long stride = (long)gridDim.x * blockDim.x;
    float4 z = make_float4(0.f, 0.f, 0.f, 0.f);
    for (; i < n4; i += stride) p[i] = z;
}

// ---------------------------------------------------------------------------
// C[M,128] = A[M,128] @ W[128,128]   (no bias, no activation)
// grid.x = M/16 (M is a multiple of 16). 256 threads = 8 waves; wave w owns
// the 16-column tile n in [16w, 16w+16).
__global__ __launch_bounds__(256) void gemm128_wmma(const float* __restrict__ A,
                                                    const float* __restrict__ W,
                                                    float* __restrict__ C) {
    const int lane  = threadIdx.x & 31;
    const int wave  = threadIdx.x >> 5;
    const int mtile = blockIdx.x;
    const int nh    = lane & 15;          // N (B/C frag) or M (A frag) sub-index
    const int koff  = (lane >> 4) * 2;    // A/B: lanes 0-15 hold K=0,1; 16-31 hold K=2,3
    const int m     = mtile * 16 + nh;
    const int n     = wave * 16 + nh;

    const float* arow = A + (size_t)m * HDIM;
    v8f acc = {};
#pragma unroll
    for (int k = 0; k < HDIM; k += 4) {
        v2f a = *(const v2f*)(arow + k + koff);
        v2f b;
        b.x = W[(size_t)(k + koff) * HDIM + n];
        b.y = W[(size_t)(k + koff + 1) * HDIM + n];
        acc = wmma_f32_k4(a, b, acc);
    }
    // C/D layout: lanes 0-15 -> rows 0..7, lanes 16-31 -> rows 8..15
    const int rowbase = mtile * 16 + (lane >> 4) * 8;
    float* crow = C + (size_t)rowbase * HDIM + n;
#pragma unroll
    for (int r = 0; r < 8; ++r) crow[(size_t)r * HDIM] = acc[r];
}

// ---------------------------------------------------------------------------
// Y[rows[e], :] += vals[e] * X[cols[e], :]   (one wave per edge, float4/lane)
__global__ __launch_bounds__(256) void spmm_atomic(const int* __restrict__ rows,
                                                   const int* __restrict__ cols,
                                                   const float* __restrict__ vals,
                                                   const float* __restrict__ X,
                                                   float* __restrict__ Y, int E) {
    const int gw   = (int)(((long)blockIdx.x * blockDim.x + threadIdx.x) >> 5);
    const int lane = threadIdx.x & 31;
    if (gw >= E) return;
    const int   r = rows[gw];
    const int   c = cols[gw];
    const float v = vals[gw];
    const float4 xv = ((const float4*)(X + (size_t)c * HDIM))[lane];
    float* y = Y + (size_t)r * HDIM + lane * 4;
    atomicAdd(y + 0, v * xv.x);
    atomicAdd(y + 1, v * xv.y);
    atomicAdd(y + 2, v * xv.z);
    atomicAdd(y + 3, v * xv.w);
}

// ---------------------------------------------------------------------------
__global__ __launch_bounds__(256) void bias_leaky(float* __restrict__ Y,
                                                  const float* __restrict__ b, long n) {
    long i = (long)blockIdx.x * blockDim.x + threadIdx.x;
    if (i >= n) return;
    float x = Y[i] + b[i & (HDIM - 1)];
    Y[i] = (x >= 0.f) ? x : ALPHA * x;
}

// ---------------------------------------------------------------------------
// mean   = [Hm | F] @ Wm + bm       (Wm: [256,128] row-major)
// logstd = [Hs | F] @ Ws + bs
// Z      = noise * exp(0.1 + 0.9*softplus(logstd)) + mean
// grid.x = M/16, 256 threads (8 waves cover N=128). Two WMMA accumulators per
// wave; phase 2 (feature half of the concat) shares the A fragment.
__global__ __launch_bounds__(256) void fused_head(const float* __restrict__ Hm,
                                                  const float* __restrict__ Hs,
                                                  const float* __restrict__ F,
                                                  const float* __restrict__ Wm,
                                                  const float* __restrict__ bm,
                                                  const float* __restrict__ Ws,
                                                  const float* __restrict__ bs,
                                                  const float* __restrict__ noise,
                                                  float* __restrict__ Z) {
    const int lane  = threadIdx.x & 31;
    const int wave  = threadIdx.x >> 5;
    const int mtile = blockIdx.x;
    const int nh    = lane & 15;
    const int koff  = (lane >> 4) * 2;
    const int m     = mtile * 16 + nh;
    const int n     = wave * 16 + nh;

    const float* hmrow = Hm + (size_t)m * HDIM;
    const float* hsrow = Hs + (size_t)m * HDIM;
    const float* frow  = F  + (size_t)m * HDIM;

    v8f accM = {};
    v8f accS = {};

    // Phase 1: K = 0..127 over Hm / Hs with top halves of Wm / Ws
#pragma unroll
    for (int k = 0; k < HDIM; k += 4) {
        v2f am = *(const v2f*)(hmrow + k + koff);
        v2f as = *(const v2f*)(hsrow + k + koff);
        v2f bM, bS;
        bM.x = Wm[(size_t)(k + koff) * HDIM + n];
        bM.y = Wm[(size_t)(k + koff + 1) * HDIM + n];
        bS.x = Ws[(size_t)(k + koff) * HDIM + n];
        bS.y = Ws[(size_t)(k + koff + 1) * HDIM + n];
        accM = wmma_f32_k4(am, bM, accM);
        accS = wmma_f32_k4(as, bS, accS);
    }
    // Phase 2: K = 128..255 over shared feature matrix F with bottom halves
#pragma unroll
    for (int k = 0; k < HDIM; k += 4) {
        v2f a = *(const v2f*)(frow + k + koff);
        const int kk = HDIM + k + koff;
        v2f bM, bS;
        bM.x = Wm[(size_t)kk * HDIM + n];
        bM.y = Wm[(size_t)(kk + 1) * HDIM + n];
        bS.x = Ws[(size_t)kk * HDIM + n];
        bS.y = Ws[(size_t)(kk + 1) * HDIM + n];
        accM = wmma_f32_k4(a, bM, accM);
        accS = wmma_f32_k4(a, bS, accS);
    }

    const int rowbase = mtile * 16 + (lane >> 4) * 8;
    const float bmv = bm[n];
    const float bsv = bs[n];
#pragma unroll
    for (int r = 0; r < 8; ++r) {
        const size_t idx = (size_t)(rowbase + r) * HDIM + n;
        float mean = accM[r] + bmv;
        float ls   = accS[r] + bsv;
        float sp   = (ls > 20.f) ? ls : log1pf(expf(ls));   // softplus
        float sig  = expf(0.1f + 0.9f * sp);
        Z[idx] = noise[idx] * sig + mean;
    }
}

// ---------------------------------------------------------------------------
extern "C" void kernel_launch(void* const* d_in, const int* in_sizes, int n_in,
                              void* d_out, int out_size, void* d_ws, size_t ws_size,
                              hipStream_t stream) {
    const float* ufea   = (const float*)d_in[0];
    const float* vfea   = (const float*)d_in[1];
    const int*   uv_r   = (const int*)d_in[2];
    const int*   uv_c   = (const int*)d_in[3];
    const float* uv_v   = (const float*)d_in[4];
    const int*   vu_r   = (const int*)d_in[5];
    const int*   vu_c   = (const int*)d_in[6];
    const float* vu_v   = (const float*)d_in[7];
    const float* noiseu = (const float*)d_in[8];
    const float* noisev = (const float*)d_in[9];
    const float* gc1_W  = (const float*)d_in[10];
    const float* gc1_b  = (const float*)d_in[11];
    const float* gc3m_W = (const float*)d_in[12];
    const float* gc3m_b = (const float*)d_in[13];
    const float* gc3s_W = (const float*)d_in[14];
    const float* gc3s_b = (const float*)d_in[15];
    const float* um_W   = (const float*)d_in[16];
    const float* um_b   = (const float*)d_in[17];
    const float* us_W   = (const float*)d_in[18];
    const float* us_b   = (const float*)d_in[19];
    const float* im_W   = (const float*)d_in[20];
    const float* im_b   = (const float*)d_in[21];
    const float* is_W   = (const float*)d_in[22];
    const float* is_b   = (const float*)d_in[23];

    const int NU = in_sizes[0] / HDIM;   // 100000
    const int NV = in_sizes[1] / HDIM;   //  50000
    const int E  = in_sizes[2];          // 1600000

    const size_t SU = (size_t)NU * HDIM;
    const size_t SV = (size_t)NV * HDIM;

    float* ws     = (float*)d_ws;
    float* big0   = ws;                  // SU floats
    float* big1   = ws + SU;             // SU floats
    float* small0 = ws + 2 * SU;         // SV floats
    float* small1 = ws + 2 * SU + SV;    // SV floats
    float* outU   = (float*)d_out;       // user_z [NU,128]
    float* outV   = outU + SU;           // item_z [NV,128]

    const dim3 blk(256);
    const dim3 gSpmm((E + 7) / 8);                    // 8 edges per 256-thread block
    const dim3 gZeroU(4096), gZeroV(4096);
    const dim3 gGemmU(NU / 16), gGemmV(NV / 16);
    const dim3 gEwU((int)((SU + 255) / 256)), gEwV((int)((SV + 255) / 256));

    // ---------------- user tower ----------------
    // tmp0 = ufea @ gc1_W
    gemm128_wmma<<<gGemmU, blk, 0, stream>>>(ufea, gc1_W, big0);
    // user_ho = leaky(VU @ tmp0 + gc1_b)            [NV,128]
    zero_kernel<<<gZeroV, blk, 0, stream>>>((float4*)small0, (long)(SV / 4));
    spmm_atomic<<<gSpmm, blk, 0, stream>>>(vu_r, vu_c, vu_v, big0, small0, E);
    bias_leaky<<<gEwV, blk, 0, stream>>>(small0, gc1_b, (long)SV);
    // uhm = leaky(UV @ (user_ho @ gc3m_W) + gc3m_b) [NU,128]
    gemm128_wmma<<<gGemmV, blk, 0, stream>>>(small0, gc3m_W, small1);
    zero_kernel<<<gZeroU, blk, 0, stream>>>((float4*)big1, (long)(SU / 4));
    spmm_atomic<<<gSpmm, blk, 0, stream>>>(uv_r, uv_c, uv_v, small1, big1, E);
    bias_leaky<<<gEwU, blk, 0, stream>>>(big1, gc3m_b, (long)SU);
    // uhs = leaky(UV @ (user_ho @ gc3s_W) + gc3s_b) [NU,128]
    gemm128_wmma<<<gGemmV, blk, 0, stream>>>(small0, gc3s_W, small1);
    zero_kernel<<<gZeroU, blk, 0, stream>>>((float4*)big0, (long)(SU / 4));
    spmm_atomic<<<gSpmm, blk, 0, stream>>>(uv_r, uv_c, uv_v, small1, big0, E);
    bias_leaky<<<gEwU, blk, 0, stream>>>(big0, gc3s_b, (long)SU);
    // user_z = reparam([uhm|ufea]@um_W+um_b, [uhs|ufea]@us_W+us_b, noise_u)
    fused_head<<<gGemmU, blk, 0, stream>>>(big1, big0, ufea, um_W, um_b,
                                           us_W, us_b, noiseu, outU);

    // ---------------- item tower ----------------
    // tmp0v = vfea @ gc1_W
    gemm128_wmma<<<gGemmV, blk, 0, stream>>>(vfea, gc1_W, small0);
    // item_ho = leaky(UV @ tmp0v + gc1_b)           [NU,128]
    zero_kernel<<<gZeroU, blk, 0, stream>>>((float4*)big1, (long)(SU / 4));
    spmm_atomic<<<gSpmm, blk, 0, stream>>>(uv_r, uv_c, uv_v, small0, big1, E);
    bias_leaky<<<gEwU, blk, 0, stream>>>(big1, gc1_b, (long)SU);
    // ihm = leaky(VU @ (item_ho @ gc3m_W) + gc3m_b) [NV,128]
    gemm128_wmma<<<gGemmU, blk, 0, stream>>>(big1, gc3m_W, big0);
    zero_kernel<<<gZeroV, blk, 0, stream>>>((float4*)small0, (long)(SV / 4));
    spmm_atomic<<<gSpmm, blk, 0, stream>>>(vu_r, vu_c, vu_v, big0, small0, E);
    bias_leaky<<<gEwV, blk, 0, stream>>>(small0, gc3m_b, (long)SV);
    // ihs = leaky(VU @ (item_ho @ gc3s_W) + gc3s_b) [NV,128]
    gemm128_wmma<<<gGemmU, blk, 0, stream>>>(big1, gc3s_W, big0);
    zero_kernel<<<gZeroV, blk, 0, stream>>>((float4*)small1, (long)(SV / 4));
    spmm_atomic<<<gSpmm, blk, 0, stream>>>(vu_r, vu_c, vu_v, big0, small1, E);
    bias_leaky<<<gEwV, blk, 0, stream>>>(small1, gc3s_b, (long)SV);
    // item_z = reparam([ihm|vfea]@im_W+im_b, [ihs|vfea]@is_W+is_b, noise_v)
    fused_head<<<gGemmV, blk, 0, stream>>>(small0, small1, vfea, im_W, im_b,
                                           is_W, is_b, noisev, outV);
}